// GATKernelLayer_49065706389979
// MI455X (gfx1250) — compile-verified
//
#include <hip/hip_runtime.h>
#include <hip/hip_bf16.h>
#include <cstdint>

#define FEAT_DIM 128

typedef __attribute__((ext_vector_type(2))) float v2f;
typedef __attribute__((ext_vector_type(8))) float v8f;

// Monotone order-preserving float <-> unsigned encoding (for atomic max on floats,
// including negatives). encode(-inf) = 0x007FFFFF > 0, so init value 0 is a safe
// lower bound for all finite scores.
__device__ __forceinline__ unsigned ordEncode(float f) {
  unsigned u = __float_as_uint(f);
  return (u & 0x80000000u) ? ~u : (u | 0x80000000u);
}
__device__ __forceinline__ float ordDecode(unsigned e) {
  unsigned u = (e & 0x80000000u) ? (e & 0x7fffffffu) : ~e;
  return __uint_as_float(u);
}

// ---------------------------------------------------------------------------
// Pass 0: init output accumulator + per-node softmax state (every call; the
// harness does not re-zero between graph replays).
// ---------------------------------------------------------------------------
__global__ void gat_init_kernel(float* __restrict__ out,
                                unsigned* __restrict__ smax_ord,
                                float* __restrict__ denom,
                                int total, int n_nodes) {
  int i = blockIdx.x * blockDim.x + threadIdx.x;
  if (i < total) out[i] = 0.0f;
  if (i < n_nodes) {
    smax_ord[i] = 0u;
    denom[i] = 0.0f;
  }
}

// ---------------------------------------------------------------------------
// Pass 1: per-edge attention scores via V_WMMA_F32_16X16X4_F32.
// One wave32 handles 16 edges. A = h_src rows (16xK), B = h_dst^T (Kx16);
// the 16x16 f32 accumulator's diagonal holds the 16 edge dot products.
// f32 A-matrix 16x4 layout: lanes 0-15 hold K={0,1}, lanes 16-31 hold K={2,3};
// B mirrors this with N = lane%16. Both reduce to the same float2 gather.
// ---------------------------------------------------------------------------
__global__ void gat_scores_wmma_kernel(const float* __restrict__ feat,
                                       const int* __restrict__ src,
                                       const int* __restrict__ dst,
                                       float* __restrict__ scores,
                                       unsigned* __restrict__ smax_ord,
                                       int E) {
  const int lane = threadIdx.x & 31;
  const int wave = blockIdx.x * (blockDim.x >> 5) + (threadIdx.x >> 5);
  const int e0 = wave * 16;
  if (e0 >= E) return;  // uniform across wave: EXEC stays all-ones for WMMA

  const int m    = lane & 15;
  const int half = lane >> 4;
  const int e    = min(e0 + m, E - 1);  // clamp tail; writes masked below

  const float* __restrict__ ps = feat + (size_t)src[e] * FEAT_DIM;
  const float* __restrict__ pd = feat + (size_t)dst[e] * FEAT_DIM;

  v8f c = {0.f, 0.f, 0.f, 0.f, 0.f, 0.f, 0.f, 0.f};
#pragma unroll
  for (int kk = 0; kk < FEAT_DIM; kk += 4) {
    const int kb = kk + 2 * half;
    const float2 av = *(const float2*)(ps + kb);
    const float2 bv = *(const float2*)(pd + kb);
    v2f a; a.x = av.x; a.y = av.y;
    v2f b; b.x = bv.x; b.y = bv.y;
    // D = A x B + C ; 8 args: (neg_a, A, neg_b, B, c_mod, C, reuse_a, reuse_b)
    c = __builtin_amdgcn_wmma_f32_16x16x4_f32(false, a, false, b, (short)0, c,
                                              false, false);
  }

  // Diagonal extraction. C layout: VGPR r -> (M=r, N=lane) for lanes 0-15 and
  // (M=r+8, N=lane-16) for lanes 16-31. Diagonal (M==N): edge r lives in c[r]
  // at lane r; edge r+8 lives in c[r] at lane r+24.
  float s = 0.0f;
#pragma unroll
  for (int r = 0; r < 8; ++r) {
    const float vlo = __shfl(c[r], r, 32);
    const float vhi = __shfl(c[r], r + 24, 32);
    if (lane == r)     s = vlo;
    if (lane == r + 8) s = vhi;
  }

  if (lane < 16 && (e0 + lane) < E) {
    const float score = s * (1.0f / 1.0f);  // / TEMPERATURE (== 1.0)
    scores[e0 + lane] = score;
    atomicMax(&smax_ord[dst[e0 + lane]], ordEncode(score));
  }
}

// ---------------------------------------------------------------------------
// Pass 2: ex = exp(score - segment_max); denom[dst] += ex. Buffer reused
// in-place (same thread reads then writes its own element).
// ---------------------------------------------------------------------------
__global__ void gat_expsum_kernel(const int* __restrict__ dst,
                                  float* __restrict__ scores_ex,
                                  const unsigned* __restrict__ smax_ord,
                                  float* __restrict__ denom, int E) {
  const int e = blockIdx.x * blockDim.x + threadIdx.x;
  if (e >= E) return;
  const int d = dst[e];
  const float mx = ordDecode(smax_ord[d]);
  const float ex = __expf(scores_ex[e] - mx);
  scores_ex[e] = ex;
  atomicAdd(&denom[d], ex);
}

// ---------------------------------------------------------------------------
// Pass 3: weighted scatter-aggregation. One wave per edge; each lane owns a
// float4 slice (32 lanes * 4 = 128 features). 4 f32 atomics per lane into L2.
// ---------------------------------------------------------------------------
__global__ void gat_aggregate_kernel(const float* __restrict__ feat,
                                     const int* __restrict__ src,
                                     const int* __restrict__ dst,
                                     const float* __restrict__ ex,
                                     const float* __restrict__ denom,
                                     float* __restrict__ out, int E) {
  const int lane = threadIdx.x & 31;
  const int e = blockIdx.x * (blockDim.x >> 5) + (threadIdx.x >> 5);
  if (e >= E) return;
  const int d = dst[e];
  const float w = ex[e] / denom[d];
  const float4 v = ((const float4*)(feat + (size_t)src[e] * FEAT_DIM))[lane];
  float* po = out + (size_t)d * FEAT_DIM + lane * 4;
  atomicAdd(po + 0, w * v.x);
  atomicAdd(po + 1, w * v.y);
  atomicAdd(po + 2, w * v.z);
  atomicAdd(po + 3, w * v.w);
}

// ---------------------------------------------------------------------------
// Pass 4: isolated nodes (denom == 0 <=> in-degree 0, since ex > 0 always)
// keep their original features.
// ---------------------------------------------------------------------------
__global__ void gat_fixup_kernel(const float* __restrict__ feat,
                                 const float* __restrict__ denom,
                                 float* __restrict__ out, int total) {
  const int i = blockIdx.x * blockDim.x + threadIdx.x;
  if (i >= total) return;
  const int n = i >> 7;  // / FEAT_DIM
  if (denom[n] == 0.0f) out[i] = feat[i];
}

extern "C" void kernel_launch(void* const* d_in, const int* in_sizes, int n_in,
                              void* d_out, int out_size, void* d_ws, size_t ws_size,
                              hipStream_t stream) {
  const float* feat = (const float*)d_in[0];
  const int* src    = (const int*)d_in[1];
  const int* dst    = (const int*)d_in[2];
  const int E       = in_sizes[1];
  const int N       = in_sizes[0] / FEAT_DIM;
  const int total   = N * FEAT_DIM;
  float* out        = (float*)d_out;

  // Workspace: [E] scores/ex | [N] smax_ord | [N] denom  (~2.8 MB)
  float*    scores   = (float*)d_ws;
  unsigned* smax_ord = (unsigned*)(scores + E);
  float*    denom    = (float*)(smax_ord + N);

  const int BLK = 256;

  // Pass 0: init
  gat_init_kernel<<<(total + BLK - 1) / BLK, BLK, 0, stream>>>(
      out, smax_ord, denom, total, N);

  // Pass 1: WMMA scores + segment max (16 edges per wave32)
  {
    const int waves   = (E + 15) / 16;
    const int threads = waves * 32;
    gat_scores_wmma_kernel<<<(threads + BLK - 1) / BLK, BLK, 0, stream>>>(
        feat, src, dst, scores, smax_ord, E);
  }

  // Pass 2: exp + denom
  gat_expsum_kernel<<<(E + BLK - 1) / BLK, BLK, 0, stream>>>(
      dst, scores, smax_ord, denom, E);

  // Pass 3: weighted scatter aggregation (one wave per edge)
  {
    const long long threads = (long long)E * 32;
    gat_aggregate_kernel<<<(int)((threads + BLK - 1) / BLK), BLK, 0, stream>>>(
        feat, src, dst, scores, denom, out, E);
  }

  // Pass 4: isolated-node pass-through
  gat_fixup_kernel<<<(total + BLK - 1) / BLK, BLK, 0, stream>>>(
      feat, denom, out, total);
}